// FeatureAttentionLayer_89335319757375
// MI455X (gfx1250) — compile-verified
//
#include <hip/hip_runtime.h>
#include <hip/hip_bf16.h>
#include <float.h>

#define NF     256
#define TOPK   32
#define WIN    128
#define EMB    128
#define BATCH  64
#define ALPHA  0.2f
#define MROWS  (BATCH * NF)   // 16384

typedef __attribute__((ext_vector_type(2))) float v2f;
typedef __attribute__((ext_vector_type(4))) float v4f;
typedef __attribute__((ext_vector_type(8))) float v8f;

__device__ __forceinline__ float lrelu(float x) { return x >= 0.f ? x : ALPHA * x; }

// ---------------------------------------------------------------------------
// Kernel 1: Wx[16384x128] = x_n[16384x128] @ W_n[128x128] + b_n, exact f32
// via V_WMMA_F32_16X16X4_F32.  One wave computes one 16x16 output tile.
// A-frag layout (16x4 f32): lanes 0-15 -> M=lane, K={kb,kb+1}; lanes 16-31 ->
// M=lane-16, K={kb+2,kb+3}  (contiguous float2 load per lane).
// B-frag (4x16) symmetric: lane half selects K rows {0,1} vs {2,3}.
// x_n is read exactly once -> NT loads so it doesn't displace W_n/Wx in L2.
// Wx is re-read by kernels 2/3 -> default (RT) stores keep it L2-resident.
// ---------------------------------------------------------------------------
__global__ __launch_bounds__(128) void gemm_wmma_f32(
    const float* __restrict__ A, const float* __restrict__ B,
    const float* __restrict__ bias, float* __restrict__ C) {
  const int wave = threadIdx.x >> 5;
  const int lane = threadIdx.x & 31;
  const int gw   = blockIdx.x * 4 + wave;      // 8192 tiles total
  const int mt   = gw >> 3;                    // 1024 M-tiles
  const int nt   = gw & 7;                     // 8 N-tiles
  const int m0   = mt * 16, n0 = nt * 16;
  const int hv   = lane >> 4;                  // lane half
  const int lm   = lane & 15;

  const float* arow = A + (m0 + lm) * 128 + 2 * hv;
  const float* bcol = B + (2 * hv) * 128 + (n0 + lm);

  v8f c = {};
  #pragma unroll 4
  for (int kb = 0; kb < 128; kb += 4) {
    v2f a = __builtin_nontemporal_load((const v2f*)(arow + kb));
    v2f b;
    b.x = bcol[kb * 128];
    b.y = bcol[kb * 128 + 128];
    // 8 args: (neg_a, A, neg_b, B, c_mod, C, reuse_a, reuse_b)
    c = __builtin_amdgcn_wmma_f32_16x16x4_f32(
        false, a, false, b, (short)0, c, false, false);
  }

  // D layout: VGPR r -> (M = m0 + r + 8*hv, N = n0 + lm)
  const float bn = bias[n0 + lm];
  float* crow = C + (m0 + 8 * hv) * 128 + (n0 + lm);
  #pragma unroll
  for (int r = 0; r < 8; ++r)
    crow[r * 128] = c[r] + bn;
}

// ---------------------------------------------------------------------------
// Kernel 2: p[r] = dot(Wx[r,:], a[0:128]);  q[r] = dot(Wx[r,:], a[128:256])
// One wave per row; float4 loads (L2 hits); wave32 shfl_xor reduction.
// ---------------------------------------------------------------------------
__global__ __launch_bounds__(128) void pq_kernel(
    const float* __restrict__ Wx, const float* __restrict__ a,
    float* __restrict__ p, float* __restrict__ q) {
  const int wave = threadIdx.x >> 5;
  const int lane = threadIdx.x & 31;
  const int row  = blockIdx.x * 4 + wave;      // 16384 rows

  const float4 w  = ((const float4*)(Wx + (size_t)row * 128))[lane];
  const float4 a0 = ((const float4*)a)[lane];
  const float4 a1 = ((const float4*)(a + 128))[lane];

  float pp = w.x * a0.x + w.y * a0.y + w.z * a0.z + w.w * a0.w;
  float qq = w.x * a1.x + w.y * a1.y + w.z * a1.z + w.w * a1.w;
  #pragma unroll
  for (int m = 16; m >= 1; m >>= 1) {
    pp += __shfl_xor(pp, m, 32);
    qq += __shfl_xor(qq, m, 32);
  }
  if (lane == 0) { p[row] = pp; q[row] = qq; }
}

// ---------------------------------------------------------------------------
// Kernel 3: per (b,n) block: build 32 logits from p/q, softmax over k,
// then h[b,n,k,:] = lrelu(att[k] * Wx[b, nbr[n,k], :])  (256 MB stream-out).
// Logit structure (from the concat+reshape in the reference):
//   k <  16 : e = p[b,n] + q[b,n]
//   k >= 16 : e = p[b, nbr[n,2k-32]] + q[b, nbr[n,2k-31]]
// then lrelu, + nan_to_num(bias), softmax.
// Output is written once, never read -> NT stores (stream past L2, keep the
// 8 MB Wx gather source resident in the 192 MB L2).
// ---------------------------------------------------------------------------
__global__ __launch_bounds__(256) void softmax_h_kernel(
    const float* __restrict__ Wx, const float* __restrict__ p,
    const float* __restrict__ q, const int* __restrict__ nbr,
    const float* __restrict__ bias_n, float* __restrict__ out) {
  const int bn = blockIdx.x;        // b*256 + n
  const int n  = bn & (NF - 1);
  const int b  = bn >> 8;

  __shared__ float s_att[TOPK];
  __shared__ int   s_nbr[TOPK];

  const int tid = threadIdx.x;
  if (tid < 32) {
    const int k = tid;
    s_nbr[k] = nbr[n * TOPK + k];

    float e;
    if (k < 16) {
      e = p[bn] + q[bn];
    } else {
      const int j0 = 2 * k - 32;
      const int r0 = nbr[n * TOPK + j0];
      const int r1 = nbr[n * TOPK + j0 + 1];
      e = p[b * NF + r0] + q[b * NF + r1];
    }
    e = lrelu(e);

    float bb = bias_n[n * TOPK + k];          // jnp.nan_to_num semantics
    if (isnan(bb))      bb = 0.f;
    else if (isinf(bb)) bb = bb > 0.f ? FLT_MAX : -FLT_MAX;
    e += bb;

    float mx = e;
    #pragma unroll
    for (int m = 16; m >= 1; m >>= 1) mx = fmaxf(mx, __shfl_xor(mx, m, 32));
    float ex = __expf(e - mx);
    float sm = ex;
    #pragma unroll
    for (int m = 16; m >= 1; m >>= 1) sm += __shfl_xor(sm, m, 32);
    s_att[k] = ex / sm;
  }
  __syncthreads();

  const v4f* wx4  = (const v4f*)Wx;
  v4f*       out4 = (v4f*)out;
  #pragma unroll
  for (int it = 0; it < 4; ++it) {
    const int i  = tid + it * 256;            // 1024 float4 per block
    const int k  = i >> 5;
    const int e4 = i & 31;
    const float att = s_att[k];
    const int   row = s_nbr[k];
    v4f v = wx4[((size_t)b * NF + row) * 32 + e4];
    v4f h;
    h.x = lrelu(v.x * att);
    h.y = lrelu(v.y * att);
    h.z = lrelu(v.z * att);
    h.w = lrelu(v.w * att);
    __builtin_nontemporal_store(h, &out4[(size_t)bn * 1024 + i]);
  }
}

// ---------------------------------------------------------------------------
extern "C" void kernel_launch(void* const* d_in, const int* in_sizes, int n_in,
                              void* d_out, int out_size, void* d_ws, size_t ws_size,
                              hipStream_t stream) {
  (void)in_sizes; (void)n_in; (void)out_size; (void)ws_size;
  const float* x_n    = (const float*)d_in[0];   // (64,256,128)
  // d_in[1] x_e, d_in[3] all_embeddings: unused by the reference output
  const int*   edge   = (const int*)  d_in[2];   // (2, 8192); row 0 = nbr
  const float* W_n    = (const float*)d_in[4];   // (128,128)
  const float* b_n    = (const float*)d_in[5];   // (128,)
  const float* a_vec  = (const float*)d_in[6];   // (256,1)
  const float* bias_n = (const float*)d_in[7];   // (256,32)
  float*       out    = (float*)d_out;           // (64,256,32,128)

  float* Wx = (float*)d_ws;                      // 16384*128 floats = 8 MB
  float* p  = Wx + (size_t)MROWS * EMB;          // 16384 floats
  float* q  = p + MROWS;                         // 16384 floats

  gemm_wmma_f32<<<2048, 128, 0, stream>>>(x_n, W_n, b_n, Wx);
  pq_kernel<<<4096, 128, 0, stream>>>(Wx, a_vec, p, q);
  softmax_h_kernel<<<MROWS, 256, 0, stream>>>(Wx, p, q, edge, bias_n, out);
}